// DecoderCell_74552042324249
// MI455X (gfx1250) — compile-verified
//
#include <hip/hip_runtime.h>
#include <math.h>

// Problem constants (from reference)
#define BB   512          // batch
#define TT   512          // encoder time
#define EE   512          // dim_enc
#define AD   128          // dim_att
#define RD   256          // dim_rnn
#define MEL  80
#define RFR  2
#define NDEP 2

typedef __attribute__((ext_vector_type(2))) float v2f;
typedef __attribute__((ext_vector_type(8))) float v8f;

// ---------------------------------------------------------------------------
// Generic fp32 WMMA GEMM: C[M,N] = A[M,K] * B + bias(+relu)
//   WTRANS=1: B = W[N,K]^T  (i.e. x @ W.T, torch Linear layout)
//   WTRANS=0: B = W[K,N]    (i.e. x @ W)
// One wave (32 lanes) computes one 16x16 tile with v_wmma_f32_16x16x4_f32.
// VGPR layouts per CDNA5 ISA 7.12.2:
//   A 16x4 : lanes 0-15 -> (M=lane, K=k+{0,1}), lanes 16-31 -> (M=lane-16, K=k+{2,3})
//   B 4x16 : vgpr0 -> K=k (lanes0-15) / K=k+2 (lanes16-31); vgpr1 -> K=k+1 / k+3
//   C 16x16: vgpr r -> (M=r, N=lane) lanes0-15 ; (M=r+8, N=lane-16) lanes16-31
// ---------------------------------------------------------------------------
template <bool WTRANS>
__global__ void gemm_wmma_f32(const float* __restrict__ Am,
                              const float* __restrict__ Wm,
                              const float* __restrict__ bias,
                              float* __restrict__ Cm,
                              int M, int N, int K, int ldc, int doRelu) {
    const int wave = threadIdx.x >> 5;
    const int lane = threadIdx.x & 31;
    const int tilesN = N >> 4;
    const int tile = blockIdx.x * 4 + wave;
    if (tile >= (M >> 4) * tilesN) return;
    const int tm = (tile / tilesN) << 4;
    const int tn = (tile % tilesN) << 4;
    const int l15 = lane & 15;
    const int hi  = lane >> 4;           // 0: K+0/1 half, 1: K+2/3 half

    v8f c = {0.f, 0.f, 0.f, 0.f, 0.f, 0.f, 0.f, 0.f};

    const float* __restrict__ arow = Am + (size_t)(tm + l15) * K;
    const float* __restrict__ wrow = WTRANS ? (Wm + (size_t)(tn + l15) * K) : Wm;

    for (int k = 0; k < K; k += 4) {
        const int ka = k + 2 * hi;
        v2f a, b;
        a.x = arow[ka];
        a.y = arow[ka + 1];
        if (WTRANS) {
            b.x = wrow[ka];
            b.y = wrow[ka + 1];
        } else {
            b.x = Wm[(size_t)ka * N + tn + l15];
            b.y = Wm[(size_t)(ka + 1) * N + tn + l15];
        }
        c = __builtin_amdgcn_wmma_f32_16x16x4_f32(false, a, false, b,
                                                  (short)0, c, false, false);
    }

    const int col = tn + l15;
    const float bv = bias ? bias[col] : 0.0f;
#pragma unroll
    for (int r = 0; r < 8; ++r) {
        const int row = tm + r + 8 * hi;
        float v = c[r] + bv;
        if (doRelu) v = fmaxf(v, 0.0f);
        Cm[(size_t)row * ldc + col] = v;
    }
}

// ---------------------------------------------------------------------------
// GRU gate combine (torch.nn.GRUCell math), biases already folded into gi/gh
// by the GEMM epilogue.  Optionally:
//   - copyto:  copy h into a strided buffer (concat of h_att into xc)
//   - resin/xcout: residual add (ResGRUCell) xc_out = res + h
// ---------------------------------------------------------------------------
__global__ void gru_combine(const float* __restrict__ gi,
                            const float* __restrict__ gh,
                            const float* __restrict__ hprev,
                            float* __restrict__ hout,
                            const float* __restrict__ resin,
                            float* __restrict__ xcout,
                            float* __restrict__ copyto,
                            int copyLd, int copyOff,
                            int H, int total) {
    int idx = blockIdx.x * blockDim.x + threadIdx.x;
    if (idx >= total) return;
    const int b = idx / H;
    const int j = idx - b * H;
    const float* gib = gi + (size_t)b * 3 * H;
    const float* ghb = gh + (size_t)b * 3 * H;
    const float r = 1.0f / (1.0f + __expf(-(gib[j]         + ghb[j])));
    const float z = 1.0f / (1.0f + __expf(-(gib[H + j]     + ghb[H + j])));
    const float n = tanhf(gib[2 * H + j] + r * ghb[2 * H + j]);
    const float h = (1.0f - z) * n + z * hprev[idx];
    hout[idx] = h;
    if (copyto) copyto[(size_t)b * copyLd + copyOff + j] = h;
    if (xcout)  xcout[idx] = resin[idx] + h;
}

// ---------------------------------------------------------------------------
// Fused attention: scores = enc[b] @ q[b]  -> softmax -> ctx = w^T enc[b]
// One 512-thread block (16 waves) per batch row.  Pass 2 re-reads enc[b]
// (1 MB working set) which should hit the 192 MB L2.
// ---------------------------------------------------------------------------
__global__ void attention_fused(const float* __restrict__ q,
                                const float* __restrict__ enc,
                                float* __restrict__ wout,
                                float* __restrict__ ctx) {
    __shared__ __align__(16) float q_s[EE];
    __shared__ float sc[TT];
    __shared__ float red[18];

    const int b    = blockIdx.x;
    const int tid  = threadIdx.x;          // 0..511
    const int wave = tid >> 5;
    const int lane = tid & 31;
    const float* __restrict__ encb = enc + (size_t)b * TT * EE;

    q_s[tid] = q[(size_t)b * EE + tid];
    __syncthreads();

    // Pass 1: one wave per row t; coalesced float4 loads, shfl reduction.
    for (int t = wave; t < TT; t += 16) {
        const float* __restrict__ row = encb + (size_t)t * EE;
        float s = 0.0f;
#pragma unroll
        for (int e = lane * 4; e < EE; e += 128) {
            const float4 ev = *(const float4*)(row + e);
            const float4 qv = *(const float4*)(q_s + e);
            s += ev.x * qv.x + ev.y * qv.y + ev.z * qv.z + ev.w * qv.w;
        }
#pragma unroll
        for (int m = 16; m >= 1; m >>= 1) s += __shfl_xor(s, m, 32);
        if (lane == 0) sc[t] = s;
    }
    __syncthreads();

    // Softmax over T=512 (one score per thread).
    const float v = sc[tid];
    float m = v;
#pragma unroll
    for (int o = 16; o >= 1; o >>= 1) m = fmaxf(m, __shfl_xor(m, o, 32));
    if (lane == 0) red[wave] = m;
    __syncthreads();
    if (tid == 0) {
        float mm = red[0];
        for (int i = 1; i < 16; ++i) mm = fmaxf(mm, red[i]);
        red[16] = mm;
    }
    __syncthreads();
    const float gmax = red[16];
    const float ex = __expf(v - gmax);
    float s = ex;
#pragma unroll
    for (int o = 16; o >= 1; o >>= 1) s += __shfl_xor(s, o, 32);
    if (lane == 0) red[wave] = s;
    __syncthreads();
    if (tid == 0) {
        float ss = red[0];
        for (int i = 1; i < 16; ++i) ss += red[i];
        red[17] = ss;
    }
    __syncthreads();
    const float w = ex / red[17];
    sc[tid] = w;
    wout[(size_t)b * TT + tid] = w;
    __syncthreads();

    // Pass 2: ctx[e] = sum_t w[t] * enc[b,t,e]; one e per thread, coalesced.
    float acc = 0.0f;
#pragma unroll 4
    for (int t = 0; t < TT; ++t) acc += sc[t] * encb[(size_t)t * EE + tid];
    ctx[(size_t)b * EE + tid] = acc;
}

// ---------------------------------------------------------------------------
extern "C" void kernel_launch(void* const* d_in, const int* in_sizes, int n_in,
                              void* d_out, int out_size, void* d_ws, size_t ws_size,
                              hipStream_t stream) {
    const float* x        = (const float*)d_in[0];   // [B, RFR*MEL] = [512,160]
    // d_in[1] 'w' (previous attention weights) is unused by the reference.
    const float* h_att    = (const float*)d_in[2];   // [B, 128]
    const float* h_dec    = (const float*)d_in[3];   // [2, B, 256]
    const float* enc      = (const float*)d_in[4];   // [B, T, E]
    const float* pre_w1   = (const float*)d_in[5];
    const float* pre_b1   = (const float*)d_in[6];
    const float* pre_w2   = (const float*)d_in[7];
    const float* pre_b2   = (const float*)d_in[8];
    const float* arnn_wih = (const float*)d_in[9];
    const float* arnn_whh = (const float*)d_in[10];
    const float* arnn_bih = (const float*)d_in[11];
    const float* arnn_bhh = (const float*)d_in[12];
    const float* att_W    = (const float*)d_in[13];  // [A, E] (x @ W)
    const float* ctx_W    = (const float*)d_in[14];  // [E, A] (x @ W)
    const float* dec_wih  = (const float*)d_in[15];  // [2, 768, 256]
    const float* dec_whh  = (const float*)d_in[16];
    const float* dec_bih  = (const float*)d_in[17];  // [2, 768]
    const float* dec_bhh  = (const float*)d_in[18];
    const float* fc_w     = (const float*)d_in[19];  // [160, 256]
    const float* fc_b     = (const float*)d_in[20];

    // Output layout (flat, return order)
    float* outp   = (float*)d_out;                           // [B,2,80] = 81920
    float* w_new  = outp + (size_t)BB * RFR * MEL;           // [B,T]    = 262144
    float* h_attn = w_new + (size_t)BB * TT;                 // [B,128]  = 65536
    float* h_decn = h_attn + (size_t)BB * AD;                // [2,B,256]= 262144

    // Workspace carve-up (floats)
    float* ws  = (float*)d_ws;
    float* p1  = ws;                    ws += (size_t)BB * AD;       // [B,128]
    float* p2  = ws;                    ws += (size_t)BB * AD;       // [B,128]
    float* gi  = ws;                    ws += (size_t)BB * 3 * RD;   // [B,768]
    float* gh  = ws;                    ws += (size_t)BB * 3 * RD;   // [B,768]
    float* qb  = ws;                    ws += (size_t)BB * EE;       // [B,512]
    float* ctx = ws;                    ws += (size_t)BB * EE;       // [B,512]
    float* xc0 = ws;                    ws += (size_t)BB * RD;       // [B,256]
    float* xc1 = ws;                    ws += (size_t)BB * RD;       // [B,256]
    float* xc2 = ws;                    ws += (size_t)BB * RD;       // [B,256]

    auto tiles  = [](int M, int N) { return ((M >> 4) * (N >> 4) + 3) / 4; };

    // 1) PreNet layer 1: p1 = relu(x @ pre_w1^T + pre_b1)     [512,160]x[160,128]
    gemm_wmma_f32<true><<<tiles(BB, AD), 128, 0, stream>>>(
        x, pre_w1, pre_b1, p1, BB, AD, RFR * MEL, AD, 1);
    // 2) PreNet layer 2: p2 = relu(p1 @ pre_w2^T + pre_b2)
    gemm_wmma_f32<true><<<tiles(BB, AD), 128, 0, stream>>>(
        p1, pre_w2, pre_b2, p2, BB, AD, AD, AD, 1);
    // 3) attention GRU gates: gi = p2 @ wih^T + bih ; gh = h_att @ whh^T + bhh
    gemm_wmma_f32<true><<<tiles(BB, 3 * AD), 128, 0, stream>>>(
        p2, arnn_wih, arnn_bih, gi, BB, 3 * AD, AD, 3 * AD, 0);
    gemm_wmma_f32<true><<<tiles(BB, 3 * AD), 128, 0, stream>>>(
        h_att, arnn_whh, arnn_bhh, gh, BB, 3 * AD, AD, 3 * AD, 0);
    // 4) h_att_new (also copied into xc0[:, 0:128])
    gru_combine<<<(BB * AD + 255) / 256, 256, 0, stream>>>(
        gi, gh, h_att, h_attn, nullptr, nullptr, xc0, RD, 0, AD, BB * AD);
    // 5) q = h_att_new @ att_W      [512,128]x[128,512]
    gemm_wmma_f32<false><<<tiles(BB, EE), 128, 0, stream>>>(
        h_attn, att_W, nullptr, qb, BB, EE, AD, EE, 0);
    // 6) fused scores -> softmax -> ctx  (bandwidth-dominant step)
    attention_fused<<<BB, 512, 0, stream>>>(qb, enc, w_new, ctx);
    // 7) att = ctx @ ctx_W, written straight into xc0[:, 128:256]
    gemm_wmma_f32<false><<<tiles(BB, AD), 128, 0, stream>>>(
        ctx, ctx_W, nullptr, xc0 + AD, BB, AD, EE, RD, 0);

    // 8) ResGRU stack (DEPTH=2)
    const float* xcin[3]  = {xc0, xc1, xc2};
    float*       xcout[3] = {xc0, xc1, xc2};
    for (int i = 0; i < NDEP; ++i) {
        const float* wih = dec_wih + (size_t)i * 3 * RD * RD;
        const float* whh = dec_whh + (size_t)i * 3 * RD * RD;
        const float* bih = dec_bih + (size_t)i * 3 * RD;
        const float* bhh = dec_bhh + (size_t)i * 3 * RD;
        const float* hp  = h_dec  + (size_t)i * BB * RD;
        float*       hn  = h_decn + (size_t)i * BB * RD;
        gemm_wmma_f32<true><<<tiles(BB, 3 * RD), 128, 0, stream>>>(
            xcin[i], wih, bih, gi, BB, 3 * RD, RD, 3 * RD, 0);
        gemm_wmma_f32<true><<<tiles(BB, 3 * RD), 128, 0, stream>>>(
            hp, whh, bhh, gh, BB, 3 * RD, RD, 3 * RD, 0);
        gru_combine<<<(BB * RD + 255) / 256, 256, 0, stream>>>(
            gi, gh, hp, hn, xcin[i], xcout[i + 1], nullptr, 0, 0, RD, BB * RD);
    }

    // 9) out = xc2 @ fc_w^T + fc_b   [512,256]x[256,160]
    gemm_wmma_f32<true><<<tiles(BB, RFR * MEL), 128, 0, stream>>>(
        xc2, fc_w, fc_b, outp, BB, RFR * MEL, RD, RFR * MEL, 0);
}